// GATLayer_38482906972428
// MI455X (gfx1250) — compile-verified
//
#include <hip/hip_runtime.h>
#include <hip/hip_bf16.h>
#include <math.h>

typedef __attribute__((ext_vector_type(2))) float v2f;
typedef __attribute__((ext_vector_type(8))) float v8f;

#define IN_DIM  256
#define OUT_DIM 128
#define KC 32          // K-chunk staged in LDS
#define NB 32          // nodes per block
#define HSTR 36        // padded LDS row stride for h tile (16B-aligned, bank-spread)

// ---------- ordered-key float <-> uint for atomic max on floats ----------
__device__ __forceinline__ unsigned fkey(float f) {
    unsigned u = __float_as_uint(f);
    return (u & 0x80000000u) ? ~u : (u | 0x80000000u);
}
__device__ __forceinline__ float keyf(unsigned k) {
    unsigned u = (k & 0x80000000u) ? (k & 0x7FFFFFFFu) : ~k;
    return __uint_as_float(u);
}

// ---------------- init: zero accumulators ----------------
__global__ void gat_init(float* __restrict__ out, unsigned* __restrict__ mkey,
                         unsigned* __restrict__ deg, float* __restrict__ denom,
                         int n, int total) {
    int i = blockIdx.x * blockDim.x + threadIdx.x;
    if (i < total) out[i] = 0.0f;
    if (i < n) { mkey[i] = 0u; deg[i] = 0u; denom[i] = 0.0f; }
}

// ---------------- z = h @ W^T via V_WMMA_F32_16X16X4_F32 ----------------
// Block: 256 threads = 8 waves. Block covers 32 nodes x 128 outs.
// Wave w: nodeTile = w>>2 (16 nodes), outGroup = w&3 (2 x 16-out tiles).
__global__ __launch_bounds__(256) void gat_gemm_wmma(
    const float* __restrict__ h, const float* __restrict__ W,
    float* __restrict__ z, int n) {
    __shared__ float lds_h[NB * HSTR];       // 32 rows x 36 floats
    __shared__ float lds_w[KC * OUT_DIM];    // pair-interleaved: ((k>>1)*128+o)*2+(k&1)

    const int tid      = threadIdx.x;
    const int lane     = tid & 31;
    const int wave     = tid >> 5;
    const int nodeTile = wave >> 2;          // 0..1
    const int outGroup = wave & 3;           // 0..3
    const int nodeBase = blockIdx.x * NB;

    const int hl  = lane & 15;
    const int khi = (lane >> 4) * 2;         // K sub-offset per ISA A layout (0 or 2)

    v8f c0 = {}; v8f c1 = {};

    for (int kc = 0; kc < IN_DIM; kc += KC) {
        __syncthreads();
        // stage h tile chunk: thread t -> node t/8, float4 at k = (t%8)*4 (coalesced)
        {
            int nodeL = tid >> 3;
            int ks    = (tid & 7) * 4;
            int g     = nodeBase + nodeL;
            if (g >= n) g = n - 1;           // clamp; bogus rows masked at store
            const float4 hv = *(const float4*)(h + (size_t)g * IN_DIM + kc + ks);
            *(float4*)(&lds_h[nodeL * HSTR + ks]) = hv;
            if (kc + KC < IN_DIM)            // CDNA5 global_prefetch of next chunk
                __builtin_prefetch((const void*)(h + (size_t)g * IN_DIM + kc + KC + ks), 0, 1);
        }
        // stage W chunk transposed + K-pair interleaved: thread t -> row o=t>>1, 16 k's
        {
            int o  = tid >> 1;
            int kb = (tid & 1) * 16;
            const float* wrow = W + (size_t)o * IN_DIM + kc + kb;
            #pragma unroll
            for (int j = 0; j < 16; ++j) {
                int k = kb + j;
                lds_w[((k >> 1) * OUT_DIM + o) * 2 + (k & 1)] = wrow[j];
            }
        }
        __syncthreads();

        const int row = nodeTile * 16 + hl;
        const int o0  = (outGroup * 2 + 0) * 16 + hl;
        const int o1  = (outGroup * 2 + 1) * 16 + hl;
        #pragma unroll
        for (int ks = 0; ks < KC; ks += 4) {
            // A fragment (16x4 f32): lane<16 holds K=ks..ks+1, lane>=16 holds K=ks+2..ks+3
            v2f a = *(const v2f*)(&lds_h[row * HSTR + ks + khi]);
            int kb = ks + khi;
            // B fragments (4x16 f32): VGPR v -> K=v (lanes 0-15) / K=v+2 (lanes 16-31)
            v2f b0 = *(const v2f*)(&lds_w[((kb >> 1) * OUT_DIM + o0) * 2]);
            v2f b1 = *(const v2f*)(&lds_w[((kb >> 1) * OUT_DIM + o1) * 2]);
            c0 = __builtin_amdgcn_wmma_f32_16x16x4_f32(false, a, false, b0, (short)0, c0, false, false);
            c1 = __builtin_amdgcn_wmma_f32_16x16x4_f32(false, a, false, b1, (short)0, c1, false, false);
        }
    }

    // D layout: VGPR r -> M=r (lanes 0-15) / M=r+8 (lanes 16-31), N = lane&15
    const int mrow0 = nodeBase + nodeTile * 16 + (lane >> 4) * 8;
    const int ncol  = lane & 15;
    #pragma unroll
    for (int r = 0; r < 8; ++r) {
        int gnode = mrow0 + r;
        if (gnode < n) {
            z[(size_t)gnode * OUT_DIM + (outGroup * 2 + 0) * 16 + ncol] = c0[r];
            z[(size_t)gnode * OUT_DIM + (outGroup * 2 + 1) * 16 + ncol] = c1[r];
        }
    }
}

// ---------------- zs = z . a_src, zd = z . a_dst (one wave per node) ----------------
__global__ __launch_bounds__(256) void gat_nodedot(
    const float* __restrict__ z, const float* __restrict__ a_attn,
    float* __restrict__ zs, float* __restrict__ zd, int n) {
    int node = blockIdx.x * 8 + (threadIdx.x >> 5);
    int lane = threadIdx.x & 31;
    if (node >= n) return;
    float4 zv = *(const float4*)(z + (size_t)node * OUT_DIM + lane * 4);
    float4 as = *(const float4*)(a_attn + lane * 4);
    float4 ad = *(const float4*)(a_attn + OUT_DIM + lane * 4);
    float s = zv.x * as.x + zv.y * as.y + zv.z * as.z + zv.w * as.w;
    float d = zv.x * ad.x + zv.y * ad.y + zv.z * ad.z + zv.w * ad.w;
    #pragma unroll
    for (int off = 16; off > 0; off >>= 1) {
        s += __shfl_down(s, off, 32);
        d += __shfl_down(d, off, 32);
    }
    if (lane == 0) { zs[node] = s; zd[node] = d; }
}

// ---------------- edge pass A: e, segment-max, degree ----------------
__global__ void gat_edgeA(const int* __restrict__ src, const int* __restrict__ dst,
                          const float* __restrict__ zs, const float* __restrict__ zd,
                          float* __restrict__ e, unsigned* __restrict__ mkey,
                          unsigned* __restrict__ deg, int E) {
    int i = blockIdx.x * blockDim.x + threadIdx.x;
    if (i >= E) return;
    int s = src[i], d = dst[i];
    float x  = zs[s] + zd[d];
    float ev = x > 0.0f ? x : 0.01f * x;       // leaky_relu(0.01)
    e[i] = ev;
    atomicMax(&mkey[d], fkey(ev));
    atomicAdd(&deg[d], 1u);
}

// ---------------- edge pass B: ex = exp(e - m[dst]), denom ----------------
__global__ void gat_edgeB(const int* __restrict__ dst, float* ebuf,
                          const unsigned* __restrict__ mkey,
                          float* __restrict__ denom, int E) {
    int i = blockIdx.x * blockDim.x + threadIdx.x;
    if (i >= E) return;
    int d = dst[i];
    float v = expf(ebuf[i] - keyf(mkey[d]));
    ebuf[i] = v;                                // ex overwrites e
    atomicAdd(&denom[d], v);
}

// ---------------- edge pass C: out[dst] += (ex/denom) * z[src] (wave per edge) ----------------
__global__ __launch_bounds__(256) void gat_edgeC(
    const int* __restrict__ src, const int* __restrict__ dst,
    const float* __restrict__ ex, const float* __restrict__ denom,
    const float* __restrict__ z, float* __restrict__ out, int E) {
    int e    = blockIdx.x * 8 + (threadIdx.x >> 5);
    int lane = threadIdx.x & 31;
    if (e >= E) return;
    int s = src[e], d = dst[e];
    float alpha = ex[e] / denom[d];
    float4 zv = *(const float4*)(z + (size_t)s * OUT_DIM + lane * 4);
    float* o = out + (size_t)d * OUT_DIM + lane * 4;
    atomicAdd(o + 0, alpha * zv.x);
    atomicAdd(o + 1, alpha * zv.y);
    atomicAdd(o + 2, alpha * zv.z);
    atomicAdd(o + 3, alpha * zv.w);
}

// ---------------- final: isolated nodes fall back to z ----------------
__global__ void gat_final(const unsigned* __restrict__ deg,
                          const float* __restrict__ z, float* __restrict__ out, int total) {
    int i = blockIdx.x * blockDim.x + threadIdx.x;
    if (i >= total) return;
    if (deg[i >> 7] == 0u) out[i] = z[i];
}

extern "C" void kernel_launch(void* const* d_in, const int* in_sizes, int n_in,
                              void* d_out, int out_size, void* d_ws, size_t ws_size,
                              hipStream_t stream) {
    const float* h      = (const float*)d_in[0];
    const int*   src    = (const int*)d_in[1];
    const int*   dst    = (const int*)d_in[2];
    const float* W      = (const float*)d_in[3];
    const float* a_attn = (const float*)d_in[4];
    const int n = in_sizes[0] / IN_DIM;     // 50000
    const int E = in_sizes[1];              // 400000
    float* out = (float*)d_out;

    // workspace carve-up (~28.2 MB total)
    char* ws = (char*)d_ws;
    float*    z     = (float*)ws;    ws += (size_t)n * OUT_DIM * sizeof(float);
    float*    zs    = (float*)ws;    ws += (size_t)n * sizeof(float);
    float*    zd    = (float*)ws;    ws += (size_t)n * sizeof(float);
    float*    ebuf  = (float*)ws;    ws += (size_t)E * sizeof(float);
    unsigned* mkey  = (unsigned*)ws; ws += (size_t)n * sizeof(unsigned);
    unsigned* deg   = (unsigned*)ws; ws += (size_t)n * sizeof(unsigned);
    float*    denom = (float*)ws;

    const int total = n * OUT_DIM;
    gat_init     <<<(total + 255) / 256, 256, 0, stream>>>(out, mkey, deg, denom, n, total);
    gat_gemm_wmma<<<(n + NB - 1) / NB,   256, 0, stream>>>(h, W, z, n);
    gat_nodedot  <<<(n + 7) / 8,         256, 0, stream>>>(z, a_attn, zs, zd, n);
    gat_edgeA    <<<(E + 255) / 256,     256, 0, stream>>>(src, dst, zs, zd, ebuf, mkey, deg, E);
    gat_edgeB    <<<(E + 255) / 256,     256, 0, stream>>>(dst, ebuf, mkey, denom, E);
    gat_edgeC    <<<(E + 7) / 8,         256, 0, stream>>>(src, dst, ebuf, denom, z, out, E);
    gat_final    <<<(total + 255) / 256, 256, 0, stream>>>(deg, z, out, total);
}